// GIN_1211180778047
// MI455X (gfx1250) — compile-verified
//
#include <hip/hip_runtime.h>
#include <math.h>

#define N_NODES   100000
#define N_EDGES   1600000
#define FEATS     64
#define N_CLASSES 40
#define N_LAYERS  3
#define BN_EPS    1e-5f

#define TILE_ROWS 128
#define ZSTRIDE   68   // padded LDS row stride (floats): 16B-aligned, bank-spread
#define WSTRIDE   68

typedef float v2f __attribute__((ext_vector_type(2)));
typedef float v8f __attribute__((ext_vector_type(8)));

__device__ __forceinline__ v8f wmma4(v2f a, v2f b, v8f c) {
  // D = A(16x4 f32) x B(4x16 f32) + C(16x16 f32), full f32 precision
  return __builtin_amdgcn_wmma_f32_16x16x4_f32(false, a, false, b, (short)0, c,
                                               false, false);
}

// gfx1250 async copy: global -> LDS, 16B per lane, tracked by ASYNCcnt.
// lds_off = low 32 bits of the generic pointer to __shared__ (the DS offset).
__device__ __forceinline__ void async_b128(const float4* g, float* lds_ptr) {
  unsigned lds_off = (unsigned)(uintptr_t)lds_ptr;
  asm volatile("global_load_async_to_lds_b128 %0, %1, off"
               :
               : "v"(lds_off), "v"(g)
               : "memory");
}
__device__ __forceinline__ void wait_async0() {
  asm volatile("s_wait_asynccnt 0x0" ::: "memory");
}

// ---------------------------------------------------------------------------
// Edge scatter: z[dst] += h[src].
// Block stages 256 edge indices in LDS (coalesced), then 16 feature-threads
// per edge do one float4 gather + 4 native f32 atomics.
// ---------------------------------------------------------------------------
__global__ __launch_bounds__(256) void scatter_add_kernel(
    const float4* __restrict__ h, float* __restrict__ z,
    const int* __restrict__ src, const int* __restrict__ dst) {
  __shared__ int sSrc[256];
  __shared__ int sDst[256];
  const int tid = threadIdx.x;
  const int e = blockIdx.x * 256 + tid;
  sSrc[tid] = (e < N_EDGES) ? src[e] : -1;
  sDst[tid] = (e < N_EDGES) ? dst[e] : -1;
  __syncthreads();

  const int fg = tid & 15;       // float4 group within 64 feats
  const int grp = tid >> 4;      // 0..15
#pragma unroll 4
  for (int i = 0; i < 16; ++i) {
    int slot = i * 16 + grp;
    int s = sSrc[slot];
    int d = sDst[slot];
    if (s >= 0) {
      float4 v = h[(unsigned)s * 16u + fg];
      float* zp = z + (unsigned)d * 64u + fg * 4u;
      unsafeAtomicAdd(zp + 0, v.x);
      unsafeAtomicAdd(zp + 1, v.y);
      unsafeAtomicAdd(zp + 2, v.z);
      unsafeAtomicAdd(zp + 3, v.w);
    }
  }
}

// ---------------------------------------------------------------------------
// Fused GIN apply_func: u = relu(relu(z@W1+b1)@W2+b2), + BN sum/sumsq atomics
// 256 threads = 8 waves; wave w owns rows [16w,16w+16) of a 128-row tile.
// ---------------------------------------------------------------------------
__global__ __launch_bounds__(256) void gin_mlp_kernel(
    const float* __restrict__ z, const float* __restrict__ W1,
    const float* __restrict__ b1, const float* __restrict__ W2,
    const float* __restrict__ b2, float* __restrict__ u,
    float* __restrict__ stats, int nrows) {
  __shared__ __align__(16) float sZ[TILE_ROWS * ZSTRIDE];
  __shared__ __align__(16) float sW[64 * WSTRIDE];
  __shared__ float sSum[64];
  __shared__ float sSq[64];

  const int tid = threadIdx.x;
  const int rowBase = blockIdx.x * TILE_ROWS;

  // async-stage z tile (zero-pad OOB rows) and W1
  for (int k = tid; k < TILE_ROWS * 16; k += 256) {
    int r = k >> 4, cg = k & 15;
    float* lp = &sZ[r * ZSTRIDE + cg * 4];
    if (rowBase + r < nrows)
      async_b128(((const float4*)z) + (rowBase + r) * 16 + cg, lp);
    else
      *(float4*)lp = make_float4(0.f, 0.f, 0.f, 0.f);
  }
  for (int k = tid; k < 64 * 16; k += 256) {
    int r = k >> 4, cg = k & 15;
    async_b128(((const float4*)W1) + k, &sW[r * WSTRIDE + cg * 4]);
  }
  if (tid < 64) { sSum[tid] = 0.f; sSq[tid] = 0.f; }
  wait_async0();
  __syncthreads();

  const int wave = tid >> 5;
  const int lane = tid & 31;
  const int lr = lane & 15;   // A: row-in-tile %16 / B,C: col %16
  const int lh = lane >> 4;   // half-wave select

  const v8f vzero = {0.f, 0.f, 0.f, 0.f, 0.f, 0.f, 0.f, 0.f};
  v8f acc[4] = {vzero, vzero, vzero, vzero};

  const float* aRow = &sZ[(wave * 16 + lr) * ZSTRIDE + 2 * lh];

  // GEMM1: t = z @ W1
#pragma unroll
  for (int kk = 0; kk < 16; ++kk) {
    v2f a = *(const v2f*)(aRow + kk * 4);
    int krow = kk * 4 + 2 * lh;
#pragma unroll
    for (int nt = 0; nt < 4; ++nt) {
      v2f b;
      b.x = sW[krow * WSTRIDE + nt * 16 + lr];
      b.y = sW[(krow + 1) * WSTRIDE + nt * 16 + lr];
      acc[nt] = wmma4(a, b, acc[nt]);
    }
  }

  // t = relu(t + b1), stored back into this wave's private sZ rows
#pragma unroll
  for (int nt = 0; nt < 4; ++nt) {
    float bb = b1[nt * 16 + lr];
#pragma unroll
    for (int v = 0; v < 8; ++v) {
      float t = fmaxf(acc[nt][v] + bb, 0.f);
      sZ[(wave * 16 + v + 8 * lh) * ZSTRIDE + nt * 16 + lr] = t;
    }
  }
  __syncthreads();  // all waves done with W1 reads
  for (int k = tid; k < 64 * 16; k += 256) {
    int r = k >> 4, cg = k & 15;
    async_b128(((const float4*)W2) + k, &sW[r * WSTRIDE + cg * 4]);
  }
  wait_async0();
  __syncthreads();

  // GEMM2: u = t @ W2
  v8f acc2[4] = {vzero, vzero, vzero, vzero};
#pragma unroll
  for (int kk = 0; kk < 16; ++kk) {
    v2f a = *(const v2f*)(aRow + kk * 4);
    int krow = kk * 4 + 2 * lh;
#pragma unroll
    for (int nt = 0; nt < 4; ++nt) {
      v2f b;
      b.x = sW[krow * WSTRIDE + nt * 16 + lr];
      b.y = sW[(krow + 1) * WSTRIDE + nt * 16 + lr];
      acc2[nt] = wmma4(a, b, acc2[nt]);
    }
  }

  // u = relu(u + b2); store + BN partial sums (valid rows only)
#pragma unroll
  for (int nt = 0; nt < 4; ++nt) {
    float bb = b2[nt * 16 + lr];
    float s = 0.f, q = 0.f;
#pragma unroll
    for (int v = 0; v < 8; ++v) {
      int grow = rowBase + wave * 16 + v + 8 * lh;
      float t = fmaxf(acc2[nt][v] + bb, 0.f);
      if (grow < nrows) {
        u[grow * 64 + nt * 16 + lr] = t;
        s += t;
        q += t * t;
      }
    }
    atomicAdd(&sSum[nt * 16 + lr], s);  // ds_add_f32
    atomicAdd(&sSq[nt * 16 + lr], q);
  }
  __syncthreads();
  if (tid < 64) {
    unsafeAtomicAdd(&stats[tid], sSum[tid]);
    unsafeAtomicAdd(&stats[64 + tid], sSq[tid]);
  }
}

// ---------------------------------------------------------------------------
// BatchNorm apply (in place): u = (u - mean) * rsqrt(var+eps) * gamma + beta
// ---------------------------------------------------------------------------
__global__ __launch_bounds__(256) void bn_apply_kernel(
    float* __restrict__ u, const float* __restrict__ stats,
    const float* __restrict__ gamma, const float* __restrict__ beta,
    int nrows) {
  unsigned i = blockIdx.x * 256u + threadIdx.x;
  if (i >= (unsigned)nrows * 16u) return;
  int cg = i & 15;
  float invN = 1.f / (float)nrows;
  float4 v = ((float4*)u)[i];
  float in[4] = {v.x, v.y, v.z, v.w};
  float o[4];
#pragma unroll
  for (int j = 0; j < 4; ++j) {
    int c = cg * 4 + j;
    float mean = stats[c] * invN;
    float var = stats[64 + c] * invN - mean * mean;
    float sc = rsqrtf(var + BN_EPS) * gamma[c];
    o[j] = (in[j] - mean) * sc + beta[c];
  }
  ((float4*)u)[i] = make_float4(o[0], o[1], o[2], o[3]);
}

// ---------------------------------------------------------------------------
// Head: out = log_softmax(relu(h@fc1+b)@fc2+b2). fc1 via WMMA, fc2 scalar.
// ---------------------------------------------------------------------------
__global__ __launch_bounds__(256) void head_kernel(
    const float* __restrict__ h, const float* __restrict__ fc1w,
    const float* __restrict__ fc1b, const float* __restrict__ fc2w,
    const float* __restrict__ fc2b, float* __restrict__ out, int nrows) {
  __shared__ __align__(16) float sH[TILE_ROWS * ZSTRIDE];
  __shared__ __align__(16) float sW[64 * WSTRIDE];
  __shared__ __align__(16) float sW2[64 * N_CLASSES];

  const int tid = threadIdx.x;
  const int rowBase = blockIdx.x * TILE_ROWS;

  for (int k = tid; k < TILE_ROWS * 16; k += 256) {
    int r = k >> 4, cg = k & 15;
    float* lp = &sH[r * ZSTRIDE + cg * 4];
    if (rowBase + r < nrows)
      async_b128(((const float4*)h) + (rowBase + r) * 16 + cg, lp);
    else
      *(float4*)lp = make_float4(0.f, 0.f, 0.f, 0.f);
  }
  for (int k = tid; k < 64 * 16; k += 256) {
    int r = k >> 4, cg = k & 15;
    async_b128(((const float4*)fc1w) + k, &sW[r * WSTRIDE + cg * 4]);
  }
  for (int k = tid; k < 64 * N_CLASSES / 4; k += 256)
    async_b128(((const float4*)fc2w) + k, &sW2[k * 4]);
  wait_async0();
  __syncthreads();

  const int wave = tid >> 5;
  const int lane = tid & 31;
  const int lr = lane & 15;
  const int lh = lane >> 4;

  const v8f vzero = {0.f, 0.f, 0.f, 0.f, 0.f, 0.f, 0.f, 0.f};
  v8f acc[4] = {vzero, vzero, vzero, vzero};
  const float* aRow = &sH[(wave * 16 + lr) * ZSTRIDE + 2 * lh];
#pragma unroll
  for (int kk = 0; kk < 16; ++kk) {
    v2f a = *(const v2f*)(aRow + kk * 4);
    int krow = kk * 4 + 2 * lh;
#pragma unroll
    for (int nt = 0; nt < 4; ++nt) {
      v2f b;
      b.x = sW[krow * WSTRIDE + nt * 16 + lr];
      b.y = sW[(krow + 1) * WSTRIDE + nt * 16 + lr];
      acc[nt] = wmma4(a, b, acc[nt]);
    }
  }
  __syncthreads();  // everyone done reading pre-GEMM sH
#pragma unroll
  for (int nt = 0; nt < 4; ++nt) {
    float bb = fc1b[nt * 16 + lr];
#pragma unroll
    for (int v = 0; v < 8; ++v) {
      float t = fmaxf(acc[nt][v] + bb, 0.f);
      sH[(wave * 16 + v + 8 * lh) * ZSTRIDE + nt * 16 + lr] = t;
    }
  }
  __syncthreads();

  if (tid < TILE_ROWS) {
    int grow = rowBase + tid;
    if (grow < nrows) {
      float logit[N_CLASSES];
#pragma unroll
      for (int c = 0; c < N_CLASSES; ++c) logit[c] = fc2b[c];
      const float* hrow = &sH[tid * ZSTRIDE];
      for (int k = 0; k < 64; ++k) {
        float hv = hrow[k];
#pragma unroll
        for (int c = 0; c < N_CLASSES; ++c)
          logit[c] = fmaf(hv, sW2[k * N_CLASSES + c], logit[c]);
      }
      float m = logit[0];
#pragma unroll
      for (int c = 1; c < N_CLASSES; ++c) m = fmaxf(m, logit[c]);
      float s = 0.f;
#pragma unroll
      for (int c = 0; c < N_CLASSES; ++c) s += expf(logit[c] - m);
      float lse = m + logf(s);
#pragma unroll
      for (int c = 0; c < N_CLASSES; ++c) out[grow * N_CLASSES + c] = logit[c] - lse;
    }
  }
}

// ---------------------------------------------------------------------------
extern "C" void kernel_launch(void* const* d_in, const int* in_sizes, int n_in,
                              void* d_out, int out_size, void* d_ws,
                              size_t ws_size, hipStream_t stream) {
  (void)in_sizes; (void)n_in; (void)out_size; (void)ws_size;
  const float* x     = (const float*)d_in[0];
  const float* W1    = (const float*)d_in[1];
  const float* b1    = (const float*)d_in[2];
  const float* W2    = (const float*)d_in[3];
  const float* b2    = (const float*)d_in[4];
  const float* gamma = (const float*)d_in[5];
  const float* beta  = (const float*)d_in[6];
  const float* fc1w  = (const float*)d_in[7];
  const float* fc1b  = (const float*)d_in[8];
  const float* fc2w  = (const float*)d_in[9];
  const float* fc2b  = (const float*)d_in[10];
  const int*   src   = (const int*)d_in[11];
  const int*   dst   = (const int*)d_in[12];

  float* zbuf  = (float*)d_ws;                 // [N,64] z = h + agg
  float* ubuf  = zbuf + (size_t)N_NODES * 64;  // [N,64] u (BN in place -> h)
  float* stats = ubuf + (size_t)N_NODES * 64;  // [3][128] sum|sumsq

  hipMemsetAsync(stats, 0, N_LAYERS * 128 * sizeof(float), stream);

  const int nBlocksTile = (N_NODES + TILE_ROWS - 1) / TILE_ROWS;
  const int nBlocksEdge = (N_EDGES + 255) / 256;
  const int nBlocksElem = (N_NODES * 16 + 255) / 256;

  const float* h = x;
  for (int i = 0; i < N_LAYERS; ++i) {
    hipMemcpyAsync(zbuf, h, (size_t)N_NODES * 64 * sizeof(float),
                   hipMemcpyDeviceToDevice, stream);
    scatter_add_kernel<<<nBlocksEdge, 256, 0, stream>>>((const float4*)h, zbuf,
                                                        src, dst);
    gin_mlp_kernel<<<nBlocksTile, 256, 0, stream>>>(
        zbuf, W1 + i * 64 * 64, b1 + i * 64, W2 + i * 64 * 64, b2 + i * 64,
        ubuf, stats + i * 128, N_NODES);
    bn_apply_kernel<<<nBlocksElem, 256, 0, stream>>>(
        ubuf, stats + i * 128, gamma + i * 64, beta + i * 64, N_NODES);
    h = ubuf;
  }
  head_kernel<<<nBlocksTile, 256, 0, stream>>>(h, fc1w, fc1b, fc2w, fc2b,
                                               (float*)d_out, N_NODES);
}